// SlopeNet_58231166599703
// MI455X (gfx1250) — compile-verified
//
#include <hip/hip_runtime.h>
#include <math.h>
#include <stdint.h>

#define DIMC   48
#define HEADS  4
#define CD     12
#define HIDDEN 127
#define BATCH  4
#define IMG_H  256
#define IMG_W  256
#define NPIX   (IMG_H * IMG_W)        // 65536 pixels per image
#define NTOT   (BATCH * NPIX)         // 262144
#define QKV_O  (3 * DIMC)             // 144
#define FI_O   (2 * HIDDEN)           // 254
#define KPAD   128                    // padded K for the fo GEMM

typedef __attribute__((ext_vector_type(2))) float v2f;
typedef __attribute__((ext_vector_type(8))) float v8f;

// ---------------------------------------------------------------------------
// CDNA5 async global->LDS staging (ASYNCcnt-tracked), with safe fallback.
// Builtin signature (from hipcc diagnostic): param0 = addrspace(1) int*
// (printed as "__device__ int *"), param1 = addrspace(3) int*, then two imms.
// ---------------------------------------------------------------------------
#if __has_builtin(__builtin_amdgcn_global_load_async_to_lds_b32)
#define HAVE_ASYNC_LDS 1
#else
#define HAVE_ASYNC_LDS 0
#endif

typedef __attribute__((address_space(1))) int gm_int;
typedef __attribute__((address_space(3))) int lds_int;

__device__ __forceinline__ void async_cp_f32(float* lds_dst, const float* g_src) {
#if HAVE_ASYNC_LDS
  __builtin_amdgcn_global_load_async_to_lds_b32(
      (gm_int*)(uintptr_t)g_src, (lds_int*)(uintptr_t)lds_dst, 0, 0);
#else
  *lds_dst = *g_src;
#endif
}

__device__ __forceinline__ void async_wait0() {
#if HAVE_ASYNC_LDS
#if __has_builtin(__builtin_amdgcn_s_wait_asynccnt)
  __builtin_amdgcn_s_wait_asynccnt(0);
#else
  asm volatile("s_wait_asynccnt 0x0" ::: "memory");
#endif
#endif
}

// ---------------------------------------------------------------------------
// WMMA f32 16x16x4:  D = A(16x4) x B(4x16) + C(16x16)
// A layout: lanes 0-15 hold M=lane; VGPR0 = K=(half?2:0), VGPR1 = K=(half?3:1)
// B layout (mirror): lanes hold N=lane&15; VGPR0 = K=(half?2:0), VGPR1 = +1
// C/D layout: acc[r] -> M = r + (half?8:0), N = lane&15
// ---------------------------------------------------------------------------
__device__ __forceinline__ v8f wmma_k4(v2f a, v2f b, v8f c) {
  return __builtin_amdgcn_wmma_f32_16x16x4_f32(false, a, false, b, (short)0, c,
                                               false, false);
}

// NT adjacent 16x16 tiles of A[M x K](row-major, lda=K) x B[K x N](row stride ldb).
// Branch-free: A row pointer is clamped (safe; garbage rows are never stored),
// K must be a multiple of 4 and all B rows readable (caller pads).
template <int K, int NT>
__device__ __forceinline__ void gemm_tiles(const float* __restrict__ A, int Mmax,
                                           int m0, const float* __restrict__ Bm,
                                           const long ldb, int n0, v8f acc[NT]) {
  const int lane = threadIdx.x & 31;
  const int hl = lane >> 4, l15 = lane & 15;
  int mrow = m0 + l15;
  mrow = (mrow < Mmax) ? mrow : (Mmax - 1);
  const float* arow = A + (long)mrow * K;
  const float* bcol = Bm + n0 + l15;
  const v8f vzero = {0.f, 0.f, 0.f, 0.f, 0.f, 0.f, 0.f, 0.f};
#pragma unroll
  for (int t = 0; t < NT; ++t) acc[t] = vzero;
#pragma unroll
  for (int kk = 0; kk < K; kk += 4) {
    const int k0 = kk + (hl << 1);
    v2f a;
    a.x = arow[k0];
    a.y = arow[k0 + 1];
    const float* b0 = bcol + (long)k0 * ldb;
    const float* b1 = b0 + ldb;
#pragma unroll
    for (int t = 0; t < NT; ++t) {
      v2f b;
      b.x = b0[t * 16];
      b.y = b1[t * 16];
      acc[t] = wmma_k4(a, b, acc[t]);
    }
  }
}

// --- fold LN weights into 1x1-conv weights; zero-pad fo to K=128 -----------
__global__ void k_prep(const float* __restrict__ qkv_w, const float* __restrict__ ln1w,
                       const float* __restrict__ ln1b,  const float* __restrict__ fi_w,
                       const float* __restrict__ ln2w,  const float* __restrict__ ln2b,
                       const float* __restrict__ fo_w,
                       float* __restrict__ Wq, float* __restrict__ rs_q, float* __restrict__ cb_q,
                       float* __restrict__ Wf, float* __restrict__ rs_f, float* __restrict__ cb_f,
                       float* __restrict__ Wo) {
  int o = blockIdx.x * blockDim.x + threadIdx.x;
  if (o < QKV_O) {
    float rs = 0.f, cb = 0.f;
    for (int c = 0; c < DIMC; ++c) {
      float wv = qkv_w[o * DIMC + c];
      float wp = wv * ln1w[c];
      Wq[o * DIMC + c] = wp;
      rs += wp;
      cb += wv * ln1b[c];
    }
    rs_q[o] = rs; cb_q[o] = cb;
  } else if (o < QKV_O + FI_O) {
    int r = o - QKV_O;
    float rs = 0.f, cb = 0.f;
    for (int c = 0; c < DIMC; ++c) {
      float wv = fi_w[r * DIMC + c];
      float wp = wv * ln2w[c];
      Wf[r * DIMC + c] = wp;
      rs += wp;
      cb += wv * ln2b[c];
    }
    rs_f[r] = rs; cb_f[r] = cb;
  } else if (o < QKV_O + FI_O + DIMC) {
    int r = o - QKV_O - FI_O;
    for (int c = 0; c < KPAD; ++c)
      Wo[r * KPAD + c] = (c < HIDDEN) ? fo_w[r * HIDDEN + c] : 0.f;
  }
}

// --- per-pixel LayerNorm stats over 48 channels ----------------------------
__global__ void k_stats(const float* __restrict__ xin, float* __restrict__ mu,
                        float* __restrict__ rsig) {
  long i = (long)blockIdx.x * blockDim.x + threadIdx.x;
  if (i >= NTOT) return;
  long b = i >> 16;
  long p = i & (NPIX - 1);
  const float* base = xin + b * (long)DIMC * NPIX + p;
  float s = 0.f, s2 = 0.f;
  for (int c = 0; c < DIMC; ++c) {
    float v = base[(long)c * NPIX];
    s += v; s2 += v * v;
  }
  float m   = s * (1.f / DIMC);
  float var = s2 * (1.f / DIMC) - m * m;
  mu[i]   = m;
  rsig[i] = rsqrtf(var + 1e-5f);
}

// --- WMMA GEMM + folded-LN epilogue: out = (W'x - mu*rowsum)*rsig + cb -----
template <int NT>
__global__ void k_gemm_ln(const float* __restrict__ Wp, int Mmax,
                          const float* __restrict__ xb,
                          const float* __restrict__ mu, const float* __restrict__ rsig,
                          const float* __restrict__ rowsum, const float* __restrict__ cbias,
                          float* __restrict__ outb) {
  const int wave = threadIdx.x >> 5, lane = threadIdx.x & 31;
  const int l15 = lane & 15, hl = lane >> 4;
  const int n0 = (blockIdx.x * (blockDim.x >> 5) + wave) * (16 * NT);
  const int m0 = blockIdx.y * 16;
  v8f acc[NT];
  gemm_tiles<DIMC, NT>(Wp, Mmax, m0, xb, NPIX, n0, acc);
#pragma unroll
  for (int t = 0; t < NT; ++t) {
    const int n = n0 + t * 16 + l15;
    const float muv = mu[n], rsv = rsig[n];
#pragma unroll
    for (int r = 0; r < 8; ++r) {
      int m = m0 + r + hl * 8;
      if (m < Mmax)
        outb[(long)m * NPIX + n] = (acc[t][r] - muv * rowsum[m]) * rsv + cbias[m];
    }
  }
}

// --- WMMA GEMM + residual add ----------------------------------------------
template <int K, int NT>
__global__ void k_gemm_res(const float* __restrict__ A, int Mmax,
                           const float* __restrict__ Bm,
                           const float* __restrict__ resid,
                           float* __restrict__ outb) {
  const int wave = threadIdx.x >> 5, lane = threadIdx.x & 31;
  const int l15 = lane & 15, hl = lane >> 4;
  const int n0 = (blockIdx.x * (blockDim.x >> 5) + wave) * (16 * NT);
  const int m0 = blockIdx.y * 16;
  v8f acc[NT];
  gemm_tiles<K, NT>(A, Mmax, m0, Bm, NPIX, n0, acc);
#pragma unroll
  for (int t = 0; t < NT; ++t) {
    const int n = n0 + t * 16 + l15;
#pragma unroll
    for (int r = 0; r < 8; ++r) {
      int m = m0 + r + hl * 8;
      if (m < Mmax)
        outb[(long)m * NPIX + n] = acc[t][r] + resid[(long)m * NPIX + n];
    }
  }
}

// --- 3x3 depthwise tap from an LDS-staged 3-row band -----------------------
__device__ __forceinline__ float dw3_lds(const float (*band)[IMG_W],
                                         const float* __restrict__ w9, int x) {
  float s = 0.f;
#pragma unroll
  for (int r = 0; r < 3; ++r) {
#pragma unroll
    for (int kx = 0; kx < 3; ++kx) {
      int xx = x + kx - 1;
      float v = ((unsigned)xx < IMG_W) ? band[r][xx] : 0.f;  // SAME zero pad in x
      s += w9[r * 3 + kx] * v;
    }
  }
  return s;
}

// depthwise conv of qkv via async LDS row staging; emits deterministic
// per-image-row sum-of-squares partials for the q/k L2 norms.
__global__ void k_dwconv_qkv(const float* __restrict__ inb, const float* __restrict__ dw,
                             float* __restrict__ outb, float* __restrict__ sqpart) {
  const int c = blockIdx.y;        // 0..143
  const int y = blockIdx.x;        // 0..255
  const int x = threadIdx.x;       // 0..255
  const float* plane = inb + (long)c * NPIX;
  __shared__ float tile[3][IMG_W];
#pragma unroll
  for (int r = 0; r < 3; ++r) {
    int yy = y + r - 1;                       // uniform across block
    if ((unsigned)yy < IMG_H) async_cp_f32(&tile[r][x], plane + yy * IMG_W + x);
    else                      tile[r][x] = 0.f;   // SAME zero pad in y
  }
  async_wait0();
  __syncthreads();
  float s = dw3_lds(tile, dw + c * 9, x);
  outb[(long)c * NPIX + y * IMG_W + x] = s;
  if (c < 2 * DIMC) {
    __shared__ float red[256];
    red[threadIdx.x] = s * s;
    __syncthreads();
    for (int off = 128; off > 0; off >>= 1) {
      if (threadIdx.x < off) red[threadIdx.x] += red[threadIdx.x + off];
      __syncthreads();
    }
    if (threadIdx.x == 0) sqpart[c * IMG_H + y] = red[0];
  }
}

__global__ void k_sqred(const float* __restrict__ sqpart, float* __restrict__ sq, int bidx) {
  int c = threadIdx.x;
  if (c >= 2 * DIMC) return;
  float s = 0.f;
  for (int y = 0; y < IMG_H; ++y) s += sqpart[c * IMG_H + y];
  sq[bidx * 2 * DIMC + c] = s;
}

// --- channel attention, stage 1: partial q.kT tiles over pixel chunks ------
__global__ void k_attn_part(const float* __restrict__ qkvd_b, float* __restrict__ part) {
  const int chunk = blockIdx.x;                 // 0..31
  const int h = blockIdx.y;                     // head
  const int wave = threadIdx.x >> 5, lane = threadIdx.x & 31;
  const int l15 = lane & 15, hl = lane >> 4;
  const float* qrow = qkvd_b + (long)(h * CD + l15) * NPIX;          // A: M=c
  const float* krow = qkvd_b + (long)(DIMC + h * CD + l15) * NPIX;   // B: N=d
  v8f acc = {0.f, 0.f, 0.f, 0.f, 0.f, 0.f, 0.f, 0.f};
  const int base = chunk * (NPIX / 32) + wave * (NPIX / 256);
#pragma unroll 4
  for (int n = 0; n < NPIX / 256; n += 4) {
    const int k0 = base + n + (hl << 1);
    v2f a, b;
    a.x = qrow[k0]; a.y = qrow[k0 + 1];
    b.x = krow[k0]; b.y = krow[k0 + 1];
    acc = wmma_k4(a, b, acc);
  }
  __shared__ float red[8 * 256];
#pragma unroll
  for (int r = 0; r < 8; ++r) red[wave * 256 + r * 32 + lane] = acc[r];
  __syncthreads();
  if (wave == 0) {
#pragma unroll
    for (int r = 0; r < 8; ++r) {
      float s = 0.f;
      for (int w2 = 0; w2 < 8; ++w2) s += red[w2 * 256 + r * 32 + lane];
      part[(h * 32 + chunk) * 256 + r * 32 + lane] = s;
    }
  }
}

// --- channel attention, stage 2: reduce, normalize, temperature, softmax ---
__global__ void k_attn_fin(const float* __restrict__ part, const float* __restrict__ sq,
                           const float* __restrict__ temp, float* __restrict__ attn,
                           int bidx) {
  const int h = blockIdx.x;
  const int tid = threadIdx.x;
  float s = 0.f;
  for (int ch = 0; ch < 32; ++ch) s += part[(h * 32 + ch) * 256 + tid];
  __shared__ float logit[16 * 16];
  const int r = tid >> 5, lane = tid & 31;
  const int c = r + ((lane >> 4) ? 8 : 0), d = lane & 15;
  if (c < CD && d < CD) {
    float nq = fmaxf(sqrtf(sq[bidx * 2 * DIMC + h * CD + c]), 1e-12f);
    float nk = fmaxf(sqrtf(sq[bidx * 2 * DIMC + DIMC + h * CD + d]), 1e-12f);
    logit[c * 16 + d] = s * temp[h] / (nq * nk);
  }
  __syncthreads();
  if (tid < CD) {
    int cc = tid;
    float mx = -1e30f;
    for (int dd = 0; dd < CD; ++dd) mx = fmaxf(mx, logit[cc * 16 + dd]);
    float ssum = 0.f, ex[CD];
    for (int dd = 0; dd < CD; ++dd) { ex[dd] = __expf(logit[cc * 16 + dd] - mx); ssum += ex[dd]; }
    float inv = 1.f / ssum;
    for (int dd = 0; dd < CD; ++dd)
      attn[(bidx * HEADS + h) * (CD * CD) + cc * CD + dd] = ex[dd] * inv;
  }
}

// --- fold projection through attention: M_b[o][h*12+d] = sum_c P[o][hc]*A[c][d]
__global__ void k_mb(const float* __restrict__ proj_w, const float* __restrict__ attn,
                     float* __restrict__ Mb, int bidx) {
  int i = blockIdx.x * blockDim.x + threadIdx.x;
  if (i >= DIMC * DIMC) return;
  int oo = i / DIMC, hd = i % DIMC;
  int h = hd / CD, d = hd % CD;
  float s = 0.f;
  for (int c = 0; c < CD; ++c)
    s += proj_w[oo * DIMC + h * CD + c] * attn[(bidx * HEADS + h) * (CD * CD) + c * CD + d];
  Mb[bidx * DIMC * DIMC + i] = s;
}

// --- FF: async-staged depthwise conv on both halves + exact-GELU gate ------
__global__ void k_dwgelu(const float* __restrict__ tb, const float* __restrict__ dw,
                         float* __restrict__ gb) {
  const int c = blockIdx.y;       // 0..126
  const int y = blockIdx.x;
  const int x = threadIdx.x;
  const float* p1 = tb + (long)c * NPIX;
  const float* p2 = tb + (long)(HIDDEN + c) * NPIX;
  __shared__ float tile[6][IMG_W];
#pragma unroll
  for (int r = 0; r < 3; ++r) {
    int yy = y + r - 1;                       // uniform across block
    if ((unsigned)yy < IMG_H) {
      async_cp_f32(&tile[r][x],     p1 + yy * IMG_W + x);
      async_cp_f32(&tile[3 + r][x], p2 + yy * IMG_W + x);
    } else {
      tile[r][x] = 0.f;
      tile[3 + r][x] = 0.f;
    }
  }
  async_wait0();
  __syncthreads();
  float d1 = dw3_lds(&tile[0], dw + c * 9, x);
  float d2 = dw3_lds(&tile[3], dw + (HIDDEN + c) * 9, x);
  float g = 0.5f * d1 * (1.f + erff(d1 * 0.70710678118654752f));
  gb[(long)c * NPIX + y * IMG_W + x] = g * d2;
}

__global__ void k_zero(float* __restrict__ p, int n) {
  int i = blockIdx.x * blockDim.x + threadIdx.x;
  if (i < n) p[i] = 0.f;
}

extern "C" void kernel_launch(void* const* d_in, const int* in_sizes, int n_in,
                              void* d_out, int out_size, void* d_ws, size_t ws_size,
                              hipStream_t stream) {
  const float* x      = (const float*)d_in[0];
  const float* ln1w   = (const float*)d_in[1];
  const float* ln1b   = (const float*)d_in[2];
  const float* temp   = (const float*)d_in[3];
  const float* qkv_w  = (const float*)d_in[4];
  const float* qkv_dw = (const float*)d_in[5];
  const float* proj_w = (const float*)d_in[6];
  const float* ln2w   = (const float*)d_in[7];
  const float* ln2b   = (const float*)d_in[8];
  const float* fi_w   = (const float*)d_in[9];
  const float* dw2    = (const float*)d_in[10];
  const float* fo_w   = (const float*)d_in[11];
  float* out = (float*)d_out;

  // workspace carving (floats); total ~191 MB
  float* W = (float*)d_ws;
  size_t o = 0;
  float* Wq     = W + o; o += (size_t)QKV_O * DIMC;
  float* Wf     = W + o; o += (size_t)FI_O * DIMC;
  float* Wo     = W + o; o += (size_t)DIMC * KPAD;
  float* rs_q   = W + o; o += 256;
  float* cb_q   = W + o; o += 256;
  float* rs_f   = W + o; o += 256;
  float* cb_f   = W + o; o += 256;
  float* sq     = W + o; o += 512;                           // 4*96 sumsq(q,k)
  float* attnb  = W + o; o += (size_t)BATCH * HEADS * CD * CD;
  float* Mb     = W + o; o += (size_t)BATCH * DIMC * DIMC;
  float* sqpart = W + o; o += (size_t)2 * DIMC * IMG_H;
  float* part   = W + o; o += (size_t)HEADS * 32 * 256;      // attn partials
  o = (o + 255) & ~(size_t)255;
  float* mu     = W + o; o += NTOT;
  float* rsig   = W + o; o += NTOT;
  float* bufA   = W + o; o += (size_t)FI_O * NPIX;           // qkv_pre / t
  float* bufB   = W + o; o += (size_t)QKV_O * NPIX;          // qkvd
  float* bufC   = W + o; o += (size_t)KPAD * NPIX;           // gated (+zero row)
  float* x2     = W + o; o += (size_t)BATCH * DIMC * NPIX;

  k_prep<<<2, 256, 0, stream>>>(qkv_w, ln1w, ln1b, fi_w, ln2w, ln2b, fo_w,
                                Wq, rs_q, cb_q, Wf, rs_f, cb_f, Wo);
  k_stats<<<NTOT / 256, 256, 0, stream>>>(x, mu, rsig);

  for (int b = 0; b < BATCH; ++b) {
    const float* xb = x + (long)b * DIMC * NPIX;
    // qkv = LN1-folded 1x1 conv  (WMMA, 9 M-tiles x 128 N-blocks)
    k_gemm_ln<4><<<dim3(NPIX / 512, QKV_O / 16), 256, 0, stream>>>(
        Wq, QKV_O, xb, mu + (long)b * NPIX, rsig + (long)b * NPIX, rs_q, cb_q, bufA);
    // depthwise 3x3 (async LDS staging) + q/k sumsq partials
    k_dwconv_qkv<<<dim3(IMG_H, QKV_O), 256, 0, stream>>>(bufA, qkv_dw, bufB, sqpart);
    k_sqred<<<1, 128, 0, stream>>>(sqpart, sq, b);
    // 12x12 channel attention per head (WMMA over 65536-pixel K, 128 blocks)
    k_attn_part<<<dim3(32, HEADS), 256, 0, stream>>>(bufB, part);
    k_attn_fin<<<HEADS, 256, 0, stream>>>(part, sq, temp, attnb, b);
    // fold proj through attn -> per-batch 48x48
    k_mb<<<(DIMC * DIMC + 255) / 256, 256, 0, stream>>>(proj_w, attnb, Mb, b);
    // x2 = x + Mb @ v   (WMMA)
    k_gemm_res<DIMC, 4><<<dim3(NPIX / 512, DIMC / 16), 256, 0, stream>>>(
        Mb + (long)b * DIMC * DIMC, DIMC,
        bufB + (long)(2 * DIMC) * NPIX, xb, x2 + (long)b * DIMC * NPIX);
  }

  k_stats<<<NTOT / 256, 256, 0, stream>>>(x2, mu, rsig);
  // zero the padding row (row 127) of the gated buffer once per call
  k_zero<<<NPIX / 256, 256, 0, stream>>>(bufC + (long)HIDDEN * NPIX, NPIX);

  for (int b = 0; b < BATCH; ++b) {
    const float* x2b = x2 + (long)b * DIMC * NPIX;
    // t = LN2-folded fi 1x1 conv  (WMMA, 254 rows -> 16 M-tiles with store guard)
    k_gemm_ln<4><<<dim3(NPIX / 512, 16), 256, 0, stream>>>(
        Wf, FI_O, x2b, mu + (long)b * NPIX, rsig + (long)b * NPIX, rs_f, cb_f, bufA);
    // depthwise 3x3 on both halves (async LDS staging) + exact-GELU gate
    k_dwgelu<<<dim3(IMG_H, HIDDEN), 256, 0, stream>>>(bufA, dw2, bufC);
    // out = x2 + fo @ g   (WMMA, K padded to 128, zero-padded A and B row)
    k_gemm_res<KPAD, 4><<<dim3(NPIX / 512, DIMC / 16), 256, 0, stream>>>(
        Wo, DIMC, bufC, x2b, out + (long)b * DIMC * NPIX);
  }
}